// STDPLearner_42039139893630
// MI455X (gfx1250) — compile-verified
//
#include <hip/hip_runtime.h>

// ---------------------------------------------------------------------------
// STDP learner for MI455X (gfx1250, wave32, WMMA).
//   Phase 1: exponential spike traces (scan over T), written to d_ws.
//   Phase 2: fused dual GEMM (K = B*T = 2048) via V_WMMA_F32_16X16X4_F32
//            + softbound epilogue  dw = 4*w*(1-w) * (0.01/16) * (M1 - M2).
// ---------------------------------------------------------------------------

constexpr int B_ = 16;
constexpr int T_ = 128;
constexpr int N_ = 2048;          // PRE_N == POST_N
constexpr int K_ = B_ * T_;       // 2048 contraction length

typedef __attribute__((ext_vector_type(2))) float v2f;   // A/B frag: 16x4 f32 -> 2 VGPRs
typedef __attribute__((ext_vector_type(8))) float v8f;   // C/D frag: 16x16 f32 -> 8 VGPRs

// ---------------------------------------------------------------------------
// Phase 1: trace[b,0,n]=0 ; trace[b,t,n]=decay*trace[b,t-1,n]+s[b,t,n], t>=1
// (spikes[0] intentionally excluded, matching the lax.scan over s[1:])
// ---------------------------------------------------------------------------
__global__ __launch_bounds__(256)
void stdp_trace_kernel(const float* __restrict__ pre,
                       const float* __restrict__ post,
                       const int*   __restrict__ dt_p,
                       float* __restrict__ preTr,
                       float* __restrict__ postTr) {
  const float decay = __expf(-(float)(*dt_p) / 20.0f);
  const int id = blockIdx.x * blockDim.x + threadIdx.x;   // over B_*N_
  if (id >= B_ * N_) return;
  const float* __restrict__ s  = (blockIdx.y == 0) ? pre   : post;
  float*       __restrict__ tr = (blockIdx.y == 0) ? preTr : postTr;
  const int b = id / N_;
  const int n = id % N_;
  const size_t base = (size_t)b * T_ * N_ + n;
  float t = 0.0f;
  tr[base] = 0.0f;
  for (int tt = 1; tt < T_; ++tt) {
    t = decay * t + s[base + (size_t)tt * N_];
    tr[base + (size_t)tt * N_] = t;
  }
}

// ---------------------------------------------------------------------------
// Phase 2: block tile 128(q) x 64(p); 8 waves, each wave: 32x32 = 2x2 WMMA
// tiles, two accumulator sets (M1 and M2). K staged through LDS in steps
// of 32 (8 K-slices of 4, 2 K per wave half each).
//
// Fragment layouts (CDNA5 ISA 7.12.2, 32-bit 16x4 A / 4x16 B / 16x16 C):
//   A: lanes 0-15 hold (M=lane, K=0|1) in v0|v1; lanes 16-31 hold K=2|3.
//   B: lanes 0-15 hold (N=lane, K=0|1);          lanes 16-31 hold K=2|3.
//   C/D: vgpr r: lanes 0-15 -> (M=r, N=lane); lanes 16-31 -> (M=r+8, N=lane-16)
// ---------------------------------------------------------------------------
__global__ __launch_bounds__(256)
void stdp_gemm_kernel(const float* __restrict__ post,    // [K_, N_] post_spikes
                      const float* __restrict__ preTr,   // [K_, N_] pre trace
                      const float* __restrict__ postTr,  // [K_, N_] post trace
                      const float* __restrict__ pre,     // [K_, N_] pre_spikes
                      const float* __restrict__ W,       // [N_, N_] weights
                      float* __restrict__ out) {         // [N_, N_] delta_w
  __shared__ float lsPost[32 * 128];
  __shared__ float lsPostTr[32 * 128];
  __shared__ float lsPreTr[32 * 64];
  __shared__ float lsPre[32 * 64];

  const int tid  = threadIdx.x;
  const int lane = tid & 31;
  const int wave = tid >> 5;
  const int half = lane >> 4;   // selects K-half {0,1} -> K offset 2*half
  const int l16  = lane & 15;

  const int q0 = blockIdx.y * 128;   // output row block (post index q)
  const int p0 = blockIdx.x * 64;    // output col block (pre index p)

  const int wm = (wave & 3) * 32;    // wave offset inside block tile (M)
  const int wn = (wave >> 2) * 32;   // wave offset inside block tile (N)

  v8f acc1[2][2];  // post x preTr   (LTP)
  v8f acc2[2][2];  // postTr x pre   (LTD)
#pragma unroll
  for (int mi = 0; mi < 2; ++mi)
#pragma unroll
    for (int ni = 0; ni < 2; ++ni) {
      acc1[mi][ni] = (v8f)(0.0f);
      acc2[mi][ni] = (v8f)(0.0f);
    }

  for (int k0 = 0; k0 < K_; k0 += 32) {
    __syncthreads();   // protect LDS reads of previous stage

    // --- global -> LDS: 32x128 tiles (post, postTr): 1024 float4, 4/thread
#pragma unroll
    for (int i = 0; i < 4; ++i) {
      const int c   = tid + 256 * i;
      const int row = c >> 5;
      const int col = (c & 31) << 2;
      const size_t g = (size_t)(k0 + row) * N_ + q0 + col;
      const float4 vA = *(const float4*)(post   + g);
      const float4 vB = *(const float4*)(postTr + g);
      *(float4*)(&lsPost[row * 128 + col])   = vA;
      *(float4*)(&lsPostTr[row * 128 + col]) = vB;
    }
    // --- global -> LDS: 32x64 tiles (preTr, pre): 512 float4, 2/thread
#pragma unroll
    for (int i = 0; i < 2; ++i) {
      const int c   = tid + 256 * i;
      const int row = c >> 4;
      const int col = (c & 15) << 2;
      const size_t g = (size_t)(k0 + row) * N_ + p0 + col;
      const float4 vA = *(const float4*)(preTr + g);
      const float4 vB = *(const float4*)(pre   + g);
      *(float4*)(&lsPreTr[row * 64 + col]) = vA;
      *(float4*)(&lsPre[row * 64 + col])   = vB;
    }
    __syncthreads();

    // --- prefetch the next K stage into cache (global_prefetch_b8)
    if (k0 + 32 < K_) {
      const int row = tid >> 5;
      const int col = (tid & 31) << 2;
      __builtin_prefetch(post   + (size_t)(k0 + 32 + row) * N_ + q0 + col, 0, 1);
      __builtin_prefetch(postTr + (size_t)(k0 + 32 + row) * N_ + q0 + col, 0, 1);
      const int row2 = tid >> 4;
      const int col2 = (tid & 15) << 2;
      if (row2 < 32) {
        __builtin_prefetch(preTr + (size_t)(k0 + 32 + row2) * N_ + p0 + col2, 0, 1);
        __builtin_prefetch(pre   + (size_t)(k0 + 32 + row2) * N_ + p0 + col2, 0, 1);
      }
    }

    // --- 8 WMMA K-slices of depth 4
#pragma unroll
    for (int kk = 0; kk < 32; kk += 4) {
      const int kb = kk + half * 2;   // this lane-half's K rows: kb, kb+1
      v2f aP[2], aT[2], bT[2], bP[2];
#pragma unroll
      for (int mi = 0; mi < 2; ++mi) {
        const int m = wm + mi * 16 + l16;
        aP[mi].x = lsPost[kb * 128 + m];
        aP[mi].y = lsPost[(kb + 1) * 128 + m];
        aT[mi].x = lsPostTr[kb * 128 + m];
        aT[mi].y = lsPostTr[(kb + 1) * 128 + m];
      }
#pragma unroll
      for (int ni = 0; ni < 2; ++ni) {
        const int n = wn + ni * 16 + l16;
        bT[ni].x = lsPreTr[kb * 64 + n];
        bT[ni].y = lsPreTr[(kb + 1) * 64 + n];
        bP[ni].x = lsPre[kb * 64 + n];
        bP[ni].y = lsPre[(kb + 1) * 64 + n];
      }
#pragma unroll
      for (int mi = 0; mi < 2; ++mi)
#pragma unroll
        for (int ni = 0; ni < 2; ++ni) {
          acc1[mi][ni] = __builtin_amdgcn_wmma_f32_16x16x4_f32(
              false, aP[mi], false, bT[ni], (short)0, acc1[mi][ni], false, false);
          acc2[mi][ni] = __builtin_amdgcn_wmma_f32_16x16x4_f32(
              false, aT[mi], false, bP[ni], (short)0, acc2[mi][ni], false, false);
        }
    }
  }

  // --- epilogue: dw = 4*w*(1-w) * (A_PLUS/B) * (M1 - M2)
  const float scale = 0.01f / 16.0f;   // A_PLUS/B == A_MINUS/B
#pragma unroll
  for (int mi = 0; mi < 2; ++mi)
#pragma unroll
    for (int ni = 0; ni < 2; ++ni) {
#pragma unroll
      for (int vr = 0; vr < 8; ++vr) {
        const int q = q0 + wm + mi * 16 + vr + 8 * half;
        const int p = p0 + wn + ni * 16 + l16;
        const size_t o = (size_t)q * N_ + p;
        const float w  = W[o];
        const float wf = 4.0f * w * (1.0f - w);
        out[o] = wf * scale * (acc1[mi][ni][vr] - acc2[mi][ni][vr]);
      }
    }
}

// ---------------------------------------------------------------------------
extern "C" void kernel_launch(void* const* d_in, const int* in_sizes, int n_in,
                              void* d_out, int out_size, void* d_ws, size_t ws_size,
                              hipStream_t stream) {
  const float* weights = (const float*)d_in[0];          // [N_, N_]
  const float* pre     = (const float*)d_in[1];          // [B_, T_, N_]
  const float* post    = (const float*)d_in[2];          // [B_, T_, N_]
  const int*   dt      = (const int*)d_in[3];            // scalar

  float* preTr  = (float*)d_ws;                          // [B_*T_, N_]
  float* postTr = preTr + (size_t)B_ * T_ * N_;          // [B_*T_, N_]

  // Phase 1: traces (blockIdx.y selects pre/post stream)
  dim3 g1((B_ * N_ + 255) / 256, 2);
  stdp_trace_kernel<<<g1, 256, 0, stream>>>(pre, post, dt, preTr, postTr);

  // Phase 2: fused dual GEMM + softbound epilogue
  dim3 g2(N_ / 64, N_ / 128);   // (p-blocks, q-blocks) = (32, 16)
  stdp_gemm_kernel<<<g2, 256, 0, stream>>>(post, preTr, postTr, pre, weights,
                                           (float*)d_out);
}